// E2E3Loss_26852135535224
// MI455X (gfx1250) — compile-verified
//
#include <hip/hip_runtime.h>
#include <hip/hip_bf16.h>
#include <stdint.h>

// ---- problem constants (from reference setup_inputs) ----
#define T_PRED 48
#define BSZ    256
#define VSZ    4096
#define KSZ    32
#define U2     64          // 2*K
#define RIDS   1024
#define NTHR   512
#define NWAVE  16

__device__ __constant__ float EPSF = 1e-9f;
#define LAM_SEL_    1.0f
#define LAM_ID_     10.0f
#define LAM_RATE_   5.0f
#define LAM_KL_     0.1f
#define LAM_ENT_    0.05f
#define LAM_SMOOTH_ 0.5f

typedef float v2f __attribute__((ext_vector_type(2)));
typedef float v4f __attribute__((ext_vector_type(4)));
typedef float v8f __attribute__((ext_vector_type(8)));
typedef int   v4i __attribute__((ext_vector_type(4)));

// ws accumulator slots:
// 0 sel_num | 1 sel_cnt | 2 id_num | 3 id_cnt | 4 rate_sum | 5 kl_sum | 6 ent_sum | 7 smooth_sum

__global__ void e2e3_init_kernel(float* acc) {
    if (threadIdx.x < 8) acc[threadIdx.x] = 0.0f;
}

// ---------------------------------------------------------------------------
// Fused loss_id + loss_kl kernel: one block per batch element b.
// Streams out_ids (nontemporal B128 loads, register-pipelined) and trg_labels
// (async global->LDS B128, double buffered, per-wave s_wait_asynccnt).
// ---------------------------------------------------------------------------
__global__ __launch_bounds__(NTHR)
void e2e3_fused_kernel(const float* __restrict__ out_ids,
                       const float* __restrict__ trg_labels,
                       const float* __restrict__ selector_probs,
                       const int*   __restrict__ candi_ids,
                       const int*   __restrict__ routes,
                       float* __restrict__ acc)
{
    const int b    = blockIdx.x;
    const int tid  = threadIdx.x;
    const int lane = tid & 31;
    const int wv   = tid >> 5;

    __shared__ __align__(16) float labbuf[2][VSZ];          // 32 KB double buffer
    __shared__ float red_dot [T_PRED][NWAVE];               // wave partials
    __shared__ float red_lsum[T_PRED][NWAVE];
    __shared__ float hist[RIDS];                            // 4 KB route histogram
    __shared__ int   s_ids[U2];
    __shared__ int   s_rep[U2];
    __shared__ float s_probs[U2];
    __shared__ float s_redA[U2];
    __shared__ float s_redB[U2];
    __shared__ float s_bc[2];

    const unsigned long long lab_base = (unsigned long long)trg_labels;
    const unsigned la0 = (unsigned)(uintptr_t)&labbuf[0][tid * 4];
    const unsigned stage_bytes = (unsigned)(VSZ * sizeof(float));

    // issue async load of trg_labels row for step t into stage st (2 x B128 per lane)
    auto issue_lab = [&](int t, int st) {
        unsigned voff0 = ((unsigned)((t + 1) * BSZ + b) * (unsigned)VSZ
                          + (unsigned)tid * 4u) * 4u;          // bytes
        unsigned voff1 = voff0 + 8192u;                        // +512 lanes * 16B
        unsigned d0 = la0 + (unsigned)st * stage_bytes;
        unsigned d1 = d0 + 8192u;
        asm volatile("global_load_async_to_lds_b128 %0, %1, %2"
                     :: "v"(d0), "v"(voff0), "s"(lab_base) : "memory");
        asm volatile("global_load_async_to_lds_b128 %0, %1, %2"
                     :: "v"(d1), "v"(voff1), "s"(lab_base) : "memory");
    };

    float dist[8];
#pragma unroll
    for (int j = 0; j < 8; ++j) dist[j] = 0.0f;

    // prime pipelines: lab row 0 async, out row 0 to registers
    issue_lab(0, 0);
    const v4f* op0 = (const v4f*)(out_ids + ((size_t)0 * BSZ + b) * VSZ);
    v4f co0 = __builtin_nontemporal_load(&op0[tid]);
    v4f co1 = __builtin_nontemporal_load(&op0[tid + NTHR]);

    for (int t = 0; t < T_PRED; ++t) {
        v4f no0 = co0, no1 = co1;
        if (t + 1 < T_PRED) {
            const v4f* opn = (const v4f*)(out_ids + ((size_t)(t + 1) * BSZ + b) * VSZ);
            no0 = __builtin_nontemporal_load(&opn[tid]);
            no1 = __builtin_nontemporal_load(&opn[tid + NTHR]);
            issue_lab(t + 1, (t + 1) & 1);
            asm volatile("s_wait_asynccnt 0x2" ::: "memory");   // row t complete
        } else {
            asm volatile("s_wait_asynccnt 0x0" ::: "memory");
        }
        const int st = t & 1;
        v4f l0 = *(const v4f*)&labbuf[st][tid * 4];
        v4f l1 = *(const v4f*)&labbuf[st][2048 + tid * 4];

        float dot = co0.x * l0.x + co0.y * l0.y + co0.z * l0.z + co0.w * l0.w
                  + co1.x * l1.x + co1.y * l1.y + co1.z * l1.z + co1.w * l1.w;
        float ls  = l0.x + l0.y + l0.z + l0.w + l1.x + l1.y + l1.z + l1.w;

        dist[0] += co0.x; dist[1] += co0.y; dist[2] += co0.z; dist[3] += co0.w;
        dist[4] += co1.x; dist[5] += co1.y; dist[6] += co1.z; dist[7] += co1.w;

#pragma unroll
        for (int o = 16; o; o >>= 1) {
            dot += __shfl_xor(dot, o, 32);
            ls  += __shfl_xor(ls,  o, 32);
        }
        if (lane == 0) { red_dot[t][wv] = dot; red_lsum[t][wv] = ls; }

        co0 = no0; co1 = no1;
    }
    __syncthreads();

    // ---- row-sum the 48x16 wave-partial matrices with V_WMMA_F32_16X16X4_F32 ----
    // D = A(16x4) * ones(4x16) + C  => every column of D holds the row sums of A.
    if (tid < 32) {
        const int m    = lane & 15;
        const int half = lane >> 4;                 // A layout: lanes16-31 hold K=2,3
        v2f bones; bones.x = 1.0f; bones.y = 1.0f;
        float idnum = 0.0f, idcnt = 0.0f;
#pragma unroll
        for (int g = 0; g < 3; ++g) {               // 3 groups of 16 rows = 48
            v8f accD = {}; v8f accL = {};
#pragma unroll
            for (int j = 0; j < 4; ++j) {           // K-chunks cover the 16 waves
                const int kb = 4 * j + 2 * half;
                v2f aD; aD.x = red_dot [16 * g + m][kb]; aD.y = red_dot [16 * g + m][kb + 1];
                v2f aL; aL.x = red_lsum[16 * g + m][kb]; aL.y = red_lsum[16 * g + m][kb + 1];
                accD = __builtin_amdgcn_wmma_f32_16x16x4_f32(false, aD, false, bones,
                                                             (short)0, accD, false, false);
                accL = __builtin_amdgcn_wmma_f32_16x16x4_f32(false, aL, false, bones,
                                                             (short)0, accL, false, false);
            }
            if (m == 0) {                           // lane 0 -> rows 16g..+7, lane 16 -> 16g+8..+15
#pragma unroll
                for (int r = 0; r < 8; ++r) {
                    float dt  = accD[r];
                    float lsv = accL[r];
                    float msk = (lsv > 0.5f) ? 1.0f : 0.0f;
                    idnum += -logf(fmaxf(dt, EPSF)) * msk;
                    idcnt += msk;
                }
            }
        }
        idnum += __shfl_xor(idnum, 16, 32);
        idcnt += __shfl_xor(idcnt, 16, 32);
        if (tid == 0) { atomicAdd(acc + 2, idnum); atomicAdd(acc + 3, idcnt); }
    }

    // ---- KL term: route histogram of dist_r, then union/dedup KL over 2K slots ----
    for (int i = tid; i < RIDS; i += NTHR) hist[i] = 0.0f;
    __syncthreads();

    const v4i* rp = (const v4i*)(routes + (size_t)b * VSZ);
    v4i r0 = rp[tid], r1 = rp[tid + NTHR];
    atomicAdd(&hist[r0.x], dist[0]); atomicAdd(&hist[r0.y], dist[1]);
    atomicAdd(&hist[r0.z], dist[2]); atomicAdd(&hist[r0.w], dist[3]);
    atomicAdd(&hist[r1.x], dist[4]); atomicAdd(&hist[r1.y], dist[5]);
    atomicAdd(&hist[r1.z], dist[6]); atomicAdd(&hist[r1.w], dist[7]);

    if (tid < U2) {
        s_ids[tid]   = candi_ids[(size_t)b * U2 + tid];
        s_probs[tid] = selector_probs[(size_t)b * U2 + tid];
    }
    __syncthreads();

    if (tid < U2) {                                  // first-occurrence representative
        int id = s_ids[tid];
        int r = 0;
        while (s_ids[r] != id) ++r;
        s_rep[tid] = r;
    }
    __syncthreads();

    float valid = 0.0f, aggu = 0.0f, PmN = 0.0f;
    if (tid < U2) {
        valid = (s_rep[tid] == tid) ? 1.0f : 0.0f;
        float Pm = 0.0f, psum = 0.0f;
#pragma unroll 4
        for (int i = 0; i < U2; ++i) {
            float p = s_probs[i];
            psum += p;
            if (s_rep[i] == tid) Pm += p;
        }
        PmN  = Pm / fmaxf(psum, EPSF);
        aggu = valid * hist[s_ids[tid]];
        s_redA[tid] = aggu;
        s_redB[tid] = valid;
    }
    __syncthreads();
    if (tid == 0) {
        float s = 0.0f, u = 0.0f;
        for (int i = 0; i < U2; ++i) { s += s_redA[i]; u += s_redB[i]; }
        s_bc[0] = s; s_bc[1] = u;
    }
    __syncthreads();
    if (tid < U2) {
        float s = s_bc[0], ucnt = s_bc[1];
        float Pb  = (s > 0.0f) ? (aggu / fmaxf(s, EPSF)) : (valid / fmaxf(ucnt, 1.0f));
        float Pbc = fmaxf(Pb, EPSF);
        s_redA[tid] = valid * Pbc * (logf(Pbc) - logf(fmaxf(PmN, EPSF)));
    }
    __syncthreads();
    if (tid == 0) {
        float klb = 0.0f;
        for (int i = 0; i < U2; ++i) klb += s_redA[i];
        atomicAdd(acc + 5, klb);
    }
}

// ---------------------------------------------------------------------------
// Selector NLL + entropy: one thread per (b, s) pair, 512 total.
// ---------------------------------------------------------------------------
__global__ void e2e3_selent_kernel(const float* __restrict__ probs,
                                   const float* __restrict__ onehot,
                                   float* __restrict__ acc)
{
    const int idx = blockIdx.x * blockDim.x + threadIdx.x;   // exactly 512 launched
    const float* p = probs  + (size_t)idx * KSZ;
    const float* o = onehot + (size_t)idx * KSZ;
    float pt = 0.0f, osum = 0.0f, ent = 0.0f;
#pragma unroll 8
    for (int k = 0; k < KSZ; ++k) {
        float pv = p[k], ov = o[k];
        pt   += pv * ov;
        osum += ov;
        float pc = fmaxf(pv, EPSF);
        ent  += -pc * logf(pc);
    }
    float msk = (osum > 0.5f) ? 1.0f : 0.0f;
    float seln = -logf(fmaxf(pt, EPSF)) * msk;
    float ente = 0.5f * ent;
#pragma unroll
    for (int ofs = 16; ofs; ofs >>= 1) {
        seln += __shfl_xor(seln, ofs, 32);
        msk  += __shfl_xor(msk,  ofs, 32);
        ente += __shfl_xor(ente, ofs, 32);
    }
    if ((threadIdx.x & 31) == 0) {
        atomicAdd(acc + 0, seln);
        atomicAdd(acc + 1, msk);
        atomicAdd(acc + 6, ente);
    }
}

// ---------------------------------------------------------------------------
// Rate L1 + smoothness: one thread per (t, b), 48*256 total.
// ---------------------------------------------------------------------------
__global__ void e2e3_rate_kernel(const float* __restrict__ out_rates,
                                 const float* __restrict__ trg_rates,
                                 const int*   __restrict__ trg_lengths,
                                 float* __restrict__ acc)
{
    const int idx = blockIdx.x * blockDim.x + threadIdx.x;   // exactly 12288 launched
    const int t = idx / BSZ, b = idx % BSZ;
    const float r  = out_rates[idx];
    const float tr = trg_rates[(t + 1) * BSZ + b];           // aligned slice [1:-1]
    float c_rate = fabsf(r - tr);
    float c_sm = 0.0f;
    if (t < T_PRED - 1) {
        int eff = trg_lengths[b] - 3;
        if (eff < 0) eff = 0;
        if (t < eff) c_sm = fabsf(out_rates[(t + 1) * BSZ + b] - r);
    }
#pragma unroll
    for (int ofs = 16; ofs; ofs >>= 1) {
        c_rate += __shfl_xor(c_rate, ofs, 32);
        c_sm   += __shfl_xor(c_sm,   ofs, 32);
    }
    if ((threadIdx.x & 31) == 0) {
        atomicAdd(acc + 4, c_rate);
        atomicAdd(acc + 7, c_sm);
    }
}

// ---------------------------------------------------------------------------
// Finalize: combine accumulators with lambdas and denominators into d_out[0].
// ---------------------------------------------------------------------------
__global__ void e2e3_final_kernel(const float* __restrict__ acc,
                                  const int*   __restrict__ trg_lengths,
                                  float* __restrict__ out)
{
    __shared__ int sred[BSZ];
    const int tid = threadIdx.x;
    sred[tid] = trg_lengths[tid] - 2;
    __syncthreads();
    for (int s = BSZ / 2; s; s >>= 1) {
        if (tid < s) sred[tid] += sred[tid + s];
        __syncthreads();
    }
    if (tid == 0) {
        float denom_rate = fmaxf(1.0f, (float)sred[0]);
        float loss = LAM_SEL_    * (acc[0] / fmaxf(acc[1], 1.0f))
                   + LAM_ID_     *  acc[2] / fmaxf(acc[3], 1.0f)
                   + LAM_RATE_   *  acc[4] / denom_rate
                   + LAM_KL_     *  acc[5] / (float)BSZ
                   + LAM_ENT_    *  acc[6] / (float)BSZ
                   + LAM_SMOOTH_ *  acc[7];
        out[0] = loss;
    }
}

extern "C" void kernel_launch(void* const* d_in, const int* in_sizes, int n_in,
                              void* d_out, int out_size, void* d_ws, size_t ws_size,
                              hipStream_t stream)
{
    const float* selector_probs = (const float*)d_in[1];
    const float* selector_oneh  = (const float*)d_in[2];
    const float* out_ids        = (const float*)d_in[3];
    const float* out_rates      = (const float*)d_in[4];
    const float* trg_labels     = (const float*)d_in[5];
    const float* trg_rates      = (const float*)d_in[6];
    const int*   trg_lengths    = (const int*)  d_in[7];
    const int*   candi_ids      = (const int*)  d_in[8];
    const int*   routes         = (const int*)  d_in[9];

    float* acc = (float*)d_ws;
    float* out = (float*)d_out;

    e2e3_init_kernel  <<<1, 32, 0, stream>>>(acc);
    e2e3_fused_kernel <<<BSZ, NTHR, 0, stream>>>(out_ids, trg_labels, selector_probs,
                                                 candi_ids, routes, acc);
    e2e3_selent_kernel<<<2, 256, 0, stream>>>(selector_probs, selector_oneh, acc);
    e2e3_rate_kernel  <<<T_PRED, 256, 0, stream>>>(out_rates, trg_rates, trg_lengths, acc);
    e2e3_final_kernel <<<1, BSZ, 0, stream>>>(acc, trg_lengths, out);
}